// PageRankTorchSparseOptimal_62405874811049
// MI455X (gfx1250) — compile-verified
//
#include <hip/hip_runtime.h>

// PageRank (sparse scatter/gather). No dense matmul => WMMA not applicable.
// CDNA5 paths used: global_atomic_add_f32 (no-return), non-temporal
// global_load_b128 index streaming, wave32 block sizing.
//
// HBM roofline: 10 iters x 256MB of int32 edge indices + 128MB degree pass
// ~= 2.7GB => ~115us at 23.3 TB/s. Node state (inv/xs/agg = 12MB) stays in
// the 192MB L2, so gathers/atomics are L2-local. NT hints on the index
// streams prevent them from thrashing the L2-resident node arrays.

#define TPB 256

// Native clang vector type (required by __builtin_nontemporal_load; HIP's
// int4 is a struct wrapper and is rejected by the builtin).
typedef int v4i __attribute__((ext_vector_type(4)));

// agg[src[e]] += xs[tgt[e]] -- the hot kernel, branch-free vector-only path
// (remainder handled by pr_edges_tail, launched only when E % 4 != 0).
// Per thread: two NT global_load_b128 (index streams), 4 L2-resident
// gathers, 4 global_atomic_add_f32 (L2-resident).
__global__ void pr_edges(const int* __restrict__ src, const int* __restrict__ tgt,
                         const float* __restrict__ xs, float* __restrict__ agg, int n4) {
  int g = blockIdx.x * blockDim.x + threadIdx.x;
  if (g < n4) {
    int i = g * 4;
    v4i s = __builtin_nontemporal_load(reinterpret_cast<const v4i*>(src + i));
    v4i t = __builtin_nontemporal_load(reinterpret_cast<const v4i*>(tgt + i));
    float v0 = xs[t.x];
    float v1 = xs[t.y];
    float v2 = xs[t.z];
    float v3 = xs[t.w];
    atomicAdd(&agg[s.x], v0);
    atomicAdd(&agg[s.y], v1);
    atomicAdd(&agg[s.z], v2);
    atomicAdd(&agg[s.w], v3);
  }
}

__global__ void pr_edges_tail(const int* __restrict__ src, const int* __restrict__ tgt,
                              const float* __restrict__ xs, float* __restrict__ agg,
                              int start, int n) {
  int j = start + blockIdx.x * blockDim.x + threadIdx.x;
  if (j < n) atomicAdd(&agg[src[j]], xs[tgt[j]]);
}

// deg[t] += 1 for every t in target_indices (vector-only; tail separate)
__global__ void pr_degree(const int* __restrict__ tgt, float* __restrict__ deg, int n4) {
  int g = blockIdx.x * blockDim.x + threadIdx.x;
  if (g < n4) {
    v4i t = __builtin_nontemporal_load(reinterpret_cast<const v4i*>(tgt + g * 4));
    atomicAdd(&deg[t.x], 1.0f);
    atomicAdd(&deg[t.y], 1.0f);
    atomicAdd(&deg[t.z], 1.0f);
    atomicAdd(&deg[t.w], 1.0f);
  }
}

__global__ void pr_degree_tail(const int* __restrict__ tgt, float* __restrict__ deg,
                               int start, int n) {
  int j = start + blockIdx.x * blockDim.x + threadIdx.x;
  if (j < n) atomicAdd(&deg[tgt[j]], 1.0f);
}

__global__ void pr_zero_f32(float* __restrict__ p, int n) {
  int i = blockIdx.x * blockDim.x + threadIdx.x;
  if (i < n) p[i] = 0.0f;
}

// inv = 1/max(deg,1); xs = x0*inv; agg = 0   (inv buffer holds deg on entry)
__global__ void pr_prep(const float* __restrict__ x0, float* __restrict__ inv,
                        float* __restrict__ xs, float* __restrict__ agg, int n) {
  int i = blockIdx.x * blockDim.x + threadIdx.x;
  if (i < n) {
    float r = 1.0f / fmaxf(inv[i], 1.0f);
    inv[i] = r;
    xs[i]  = x0[i] * r;
    agg[i] = 0.0f;
  }
}

// xs = (agg*D + teleport) * inv ; agg = 0   (fused finalize + next-iter scale)
__global__ void pr_finish_iter(const float* __restrict__ inv, float* __restrict__ xs,
                               float* __restrict__ agg, float dfac, float tele, int n) {
  int i = blockIdx.x * blockDim.x + threadIdx.x;
  if (i < n) {
    float v = fmaf(agg[i], dfac, tele);
    xs[i]  = v * inv[i];
    agg[i] = 0.0f;
  }
}

// out = agg*D + teleport   (last iteration)
__global__ void pr_finish_last(const float* __restrict__ agg, float* __restrict__ out,
                               float dfac, float tele, int n) {
  int i = blockIdx.x * blockDim.x + threadIdx.x;
  if (i < n) out[i] = fmaf(agg[i], dfac, tele);
}

extern "C" void kernel_launch(void* const* d_in, const int* in_sizes, int n_in,
                              void* d_out, int out_size, void* d_ws, size_t ws_size,
                              hipStream_t stream) {
  const float* x0  = (const float*)d_in[0];  // node_influence, float32 [N]
  const int*   src = (const int*)d_in[1];    // source_indices, int32   [E]
  const int*   tgt = (const int*)d_in[2];    // target_indices, int32   [E]
  float*       out = (float*)d_out;

  const int N = in_sizes[0];
  const int E = in_sizes[1];

  // workspace layout: [inv | xs | agg], 3 * N floats = 12 MB
  float* inv = (float*)d_ws;
  float* xs  = inv + N;
  float* agg = xs + N;

  const float D    = 0.85f;
  const float tele = (1.0f - D) / (float)N;
  const int   NUM_ITER = 10;

  const int E4  = E / 4;       // full vec4 groups
  const int rem = E - E4 * 4;  // 0 for E = 32M

  const int nodeBlocks = (N + TPB - 1) / TPB;
  const int edgeBlocks = (E4 + TPB - 1) / TPB;

  // degree pass (once): inv <- deg counts
  pr_zero_f32<<<nodeBlocks, TPB, 0, stream>>>(inv, N);
  if (E4 > 0) pr_degree<<<edgeBlocks, TPB, 0, stream>>>(tgt, inv, E4);
  if (rem > 0) pr_degree_tail<<<1, TPB, 0, stream>>>(tgt, inv, E4 * 4, E);
  pr_prep<<<nodeBlocks, TPB, 0, stream>>>(x0, inv, xs, agg, N);

  for (int it = 0; it < NUM_ITER; ++it) {
    if (E4 > 0) pr_edges<<<edgeBlocks, TPB, 0, stream>>>(src, tgt, xs, agg, E4);
    if (rem > 0) pr_edges_tail<<<1, TPB, 0, stream>>>(src, tgt, xs, agg, E4 * 4, E);
    if (it < NUM_ITER - 1) {
      pr_finish_iter<<<nodeBlocks, TPB, 0, stream>>>(inv, xs, agg, D, tele, N);
    } else {
      pr_finish_last<<<nodeBlocks, TPB, 0, stream>>>(agg, out, D, tele, N);
    }
  }
}